// SelfAttention_18330920419849
// MI455X (gfx1250) — compile-verified
//
#include <hip/hip_runtime.h>
#include <hip/hip_bf16.h>
#include <cstddef>

typedef __attribute__((ext_vector_type(2))) float v2f;
typedef __attribute__((ext_vector_type(8))) float v8f;

#define B_  4
#define T_  2048
#define D_  2048
#define H_  16
#define HD_ 128

__device__ __forceinline__ v8f wmma_f32(v2f a, v2f b, v8f c) {
    // V_WMMA_F32_16X16X4_F32 : D(16x16 f32) = A(16x4 f32) x B(4x16 f32) + C
    return __builtin_amdgcn_wmma_f32_16x16x4_f32(false, a, false, b, (short)0, c,
                                                 false, false);
}

// ---------------------------------------------------------------------------
// Shared GEMM core: wave tile 64x64 (4x4 fp32-WMMA accumulators), block tile
// 128x256 with 8 waves (2x4). 16 WMMAs per 8 b64 loads (2x the intensity of a
// 32x32 tile); K loop = outer 64 / inner 4 with global_prefetch one outer
// step ahead.
// ---------------------------------------------------------------------------
__device__ __forceinline__ void gemm_core(const float* __restrict__ A,
                                          const float* __restrict__ W,
                                          int m0, int n0, int l16, int half,
                                          v8f acc[4][4]) {
    const int K_ = D_;
    const float* arow[4];
    const float* brow[4];
#pragma unroll
    for (int i = 0; i < 4; ++i) {
        arow[i] = A + (size_t)(m0 + i * 16 + l16) * K_ + half * 2;
        brow[i] = W + (size_t)(n0 + i * 16 + l16) * K_ + half * 2;
    }
    for (int k0 = 0; k0 < K_; k0 += 64) {
#pragma unroll
        for (int i = 0; i < 4; ++i) {               // prefetch next outer step
            __builtin_prefetch(arow[i] + k0 + 64, 0, 0);
            __builtin_prefetch(brow[i] + k0 + 64, 0, 0);
        }
#pragma unroll 2
        for (int kk = k0; kk < k0 + 64; kk += 4) {
            v2f af[4], bf[4];
#pragma unroll
            for (int i = 0; i < 4; ++i) af[i] = *(const v2f*)(arow[i] + kk);
#pragma unroll
            for (int j = 0; j < 4; ++j) bf[j] = *(const v2f*)(brow[j] + kk);
#pragma unroll
            for (int i = 0; i < 4; ++i)
#pragma unroll
                for (int j = 0; j < 4; ++j)
                    acc[i][j] = wmma_f32(af[i], bf[j], acc[i][j]);
        }
    }
}

// ---------------------------------------------------------------------------
// Kernel 1: qkv = x @ Wqkv^T, scattered into (B,H,T,hd) Q/K/V buffers.
// grid = (6144/256, 8192/128), block = 256.
// ---------------------------------------------------------------------------
__global__ __launch_bounds__(256)
void gemm_qkv(const float* __restrict__ A, const float* __restrict__ W,
              float* __restrict__ Qr, float* __restrict__ Kr, float* __restrict__ Vr) {
    const int lane = threadIdx.x & 31;
    const int wid  = threadIdx.x >> 5;             // 0..7
    const int half = lane >> 4;                    // 0/1 : K-pair select
    const int l16  = lane & 15;
    const int m0 = blockIdx.y * 128 + (wid >> 2) * 64;
    const int n0 = blockIdx.x * 256 + (wid & 3) * 64;

    v8f acc[4][4] = {};
    gemm_core(A, W, m0, n0, l16, half, acc);

    // Scatter: column n in [0,6144): n>>11 selects Q/K/V, head = (n&2047)>>7, d = n&127
#pragma unroll
    for (int j = 0; j < 4; ++j) {
        const int coln  = n0 + j * 16 + l16;
        const int which = coln >> 11;
        const int r     = coln & (D_ - 1);
        const int hh    = r >> 7;
        const int dd    = r & (HD_ - 1);
        float* dst = (which == 0) ? Qr : ((which == 1) ? Kr : Vr);
#pragma unroll
        for (int i = 0; i < 4; ++i) {
#pragma unroll
            for (int e = 0; e < 8; ++e) {
                const int row = m0 + i * 16 + e + (half << 3);
                const int bb  = row >> 11;          // row / T_
                const int t   = row & (T_ - 1);
                dst[(((size_t)(bb * H_ + hh)) * T_ + t) * HD_ + dd] = acc[i][j][e];
            }
        }
    }
}

// ---------------------------------------------------------------------------
// Kernel 2: in-place NeoX RoPE on Q and K ((B,H,T,hd) layout).
// ---------------------------------------------------------------------------
__global__ void rope_kernel(float* __restrict__ Q, float* __restrict__ K) {
    const int perTensor = B_ * H_ * T_ * (HD_ / 2);
    int idx = blockIdx.x * blockDim.x + threadIdx.x;
    if (idx >= 2 * perTensor) return;
    float* P = (idx < perTensor) ? Q : K;
    int e = (idx < perTensor) ? idx : idx - perTensor;
    const int d  = e & (HD_ / 2 - 1);
    const int t  = (e >> 6) & (T_ - 1);
    const int bh = e >> 17;                        // e / (64*2048)
    float* p = P + (((size_t)bh * T_) + t) * HD_;
    const float inv = __powf(10000.0f, -((float)(2 * d)) * (1.0f / (float)HD_));
    const float ang = (float)t * inv;
    const float s = __sinf(ang), c = __cosf(ang);
    const float q0 = p[d], q1 = p[d + HD_ / 2];
    p[d]           = q0 * c - q1 * s;
    p[d + HD_ / 2] = q1 * c + q0 * s;
}

// ---------------------------------------------------------------------------
// Kernel 3: flash attention. Block = 8 waves; each wave owns a 16-row Q tile,
// streams 16-wide K/V tiles with online softmax. Causal loop bound. P staged
// through per-wave LDS (C-layout -> A-layout) for the P@V WMMA.
// grid = (T/128, B*H)
// ---------------------------------------------------------------------------
__global__ __launch_bounds__(256)
void flash_attn(const float* __restrict__ Q, const float* __restrict__ K,
                const float* __restrict__ V, const int* __restrict__ amask,
                float* __restrict__ attn) {
    __shared__ float Pst[8][16][17];               // +1 pad: no bank conflicts
    const int lane = threadIdx.x & 31;
    const int wid  = threadIdx.x >> 5;
    const int half = lane >> 4;
    const int l16  = lane & 15;
    const int bh = blockIdx.y;                     // 0..63
    const int b  = bh >> 4;
    const int h  = bh & (H_ - 1);
    const int q0 = blockIdx.x * 128 + wid * 16;
    const float scale = 0.088388347648318447f;     // 1/sqrt(128)

    const float* Qp = Q + ((size_t)bh * T_) * HD_;
    const float* Kp = K + ((size_t)bh * T_) * HD_;
    const float* Vp = V + ((size_t)bh * T_) * HD_;

    // Preload the wave's Q tile (16x128) as 32 A-fragments (registers).
    v2f qa[32];
    {
        const float* qrow = Qp + (size_t)(q0 + l16) * HD_ + half * 2;
#pragma unroll
        for (int c = 0; c < 32; ++c) qa[c] = *(const v2f*)(qrow + c * 4);
    }

    v8f o[8] = {};
    float mrow[8], lrow[8];
#pragma unroll
    for (int i = 0; i < 8; ++i) { mrow[i] = -1e30f; lrow[i] = 0.0f; }

    const int kt_end = q0 + 16;                    // causal: keys <= last query row
    for (int kt = 0; kt < kt_end; kt += 16) {
        // ---- S = Q_tile @ K_tile^T  (16x16, fp32) ----
        v8f s = {};
        {
            const float* krow = Kp + (size_t)(kt + l16) * HD_ + half * 2;
#pragma unroll
            for (int c = 0; c < 32; ++c) {
                v2f bf = *(const v2f*)(krow + c * 4);
                s = wmma_f32(qa[c], bf, s);
            }
        }
        // ---- mask + online softmax (rows live across 16 lanes of each half) ----
        const int kcol   = kt + l16;
        const int kvalid = amask[b * T_ + kcol];
#pragma unroll
        for (int i = 0; i < 8; ++i) {
            const int qrow = q0 + i + (half << 3);
            float val = ((kcol <= qrow) && kvalid) ? s[i] * scale : -1e30f;
            float vmax = val;
#pragma unroll
            for (int off = 1; off < 16; off <<= 1)
                vmax = fmaxf(vmax, __shfl_xor(vmax, off, 16));
            const float mnew  = fmaxf(mrow[i], vmax);
            const float alpha = __expf(mrow[i] - mnew);
            const float p     = __expf(val - mnew);
            float psum = p;
#pragma unroll
            for (int off = 1; off < 16; off <<= 1)
                psum += __shfl_xor(psum, off, 16);
            lrow[i] = lrow[i] * alpha + psum;
            mrow[i] = mnew;
#pragma unroll
            for (int j = 0; j < 8; ++j) o[j][i] *= alpha;
            Pst[wid][i + (half << 3)][l16] = p;    // C-layout -> LDS
        }
        asm volatile("s_wait_dscnt 0" ::: "memory");  // P stores visible to this wave
        // ---- O += P @ V_tile  (A-frags from LDS, B-frags strided from V) ----
#pragma unroll
        for (int j = 0; j < 8; ++j) {
            const float* vb = Vp + (size_t)kt * HD_ + j * 16 + l16;
#pragma unroll
            for (int c = 0; c < 4; ++c) {
                const int kb = c * 4 + half * 2;
                v2f af; af.x = Pst[wid][l16][kb];  af.y = Pst[wid][l16][kb + 1];
                v2f bf; bf.x = vb[(size_t)kb * HD_]; bf.y = vb[(size_t)(kb + 1) * HD_];
                o[j] = wmma_f32(af, bf, o[j]);
            }
        }
    }

    // ---- normalize and write (B*T, D) row-major ----
#pragma unroll
    for (int i = 0; i < 8; ++i) {
        const float inv = 1.0f / lrow[i];
        const int qrow = q0 + i + (half << 3);
        float* orow = attn + ((size_t)(b * T_ + qrow)) * D_ + h * HD_ + l16;
#pragma unroll
        for (int j = 0; j < 8; ++j) orow[j * 16] = o[j][i] * inv;
    }
}

// ---------------------------------------------------------------------------
// Kernel 4: out = attn @ Wproj^T  (8192x2048 @ 2048x2048^T), plain store.
// grid = (2048/256, 8192/128), block = 256.
// ---------------------------------------------------------------------------
__global__ __launch_bounds__(256)
void gemm_proj(const float* __restrict__ A, const float* __restrict__ W,
               float* __restrict__ C) {
    const int lane = threadIdx.x & 31;
    const int wid  = threadIdx.x >> 5;
    const int half = lane >> 4;
    const int l16  = lane & 15;
    const int m0 = blockIdx.y * 128 + (wid >> 2) * 64;
    const int n0 = blockIdx.x * 256 + (wid & 3) * 64;

    v8f acc[4][4] = {};
    gemm_core(A, W, m0, n0, l16, half, acc);

#pragma unroll
    for (int j = 0; j < 4; ++j) {
        const int coln = n0 + j * 16 + l16;
#pragma unroll
        for (int i = 0; i < 4; ++i) {
#pragma unroll
            for (int e = 0; e < 8; ++e) {
                const int row = m0 + i * 16 + e + (half << 3);
                C[(size_t)row * D_ + coln] = acc[i][j][e];
            }
        }
    }
}

// ---------------------------------------------------------------------------
extern "C" void kernel_launch(void* const* d_in, const int* in_sizes, int n_in,
                              void* d_out, int out_size, void* d_ws, size_t ws_size,
                              hipStream_t stream) {
    (void)in_sizes; (void)n_in; (void)out_size; (void)ws_size;
    const float* x     = (const float*)d_in[0];   // (4,2048,2048)
    const float* Wqkv  = (const float*)d_in[1];   // (6144,2048)
    const float* Wproj = (const float*)d_in[2];   // (2048,2048)
    const int*   amask = (const int*)d_in[3];     // (4,2048)
    float* out = (float*)d_out;                   // (4,2048,2048)

    // workspace: Q | K | V ((B,H,T,hd) each, 64MB) | attn ((B*T,D), 64MB)
    const size_t SEG = (size_t)B_ * H_ * T_ * HD_;       // 16.78M floats
    char* ws = (char*)d_ws;
    float* Qr   = (float*)(ws);
    float* Kr   = (float*)(ws + SEG * sizeof(float));
    float* Vr   = (float*)(ws + 2 * SEG * sizeof(float));
    float* attn = (float*)(ws + 3 * SEG * sizeof(float));

    dim3 blk(256);
    // 1) QKV projection: M=8192, N=6144, K=2048
    gemm_qkv<<<dim3((3 * D_) / 256, (B_ * T_) / 128), blk, 0, stream>>>(
        x, Wqkv, Qr, Kr, Vr);
    // 2) RoPE in place on Q, K
    {
        int total = 2 * B_ * H_ * T_ * (HD_ / 2);
        rope_kernel<<<(total + 255) / 256, 256, 0, stream>>>(Qr, Kr);
    }
    // 3) flash attention -> attn (B*T, D)
    flash_attn<<<dim3(T_ / 128, B_ * H_), blk, 0, stream>>>(Qr, Kr, Vr, amask, attn);
    // 4) output projection -> d_out
    gemm_proj<<<dim3(D_ / 128 / 2, (B_ * T_) / 128), blk, 0, stream>>>(attn, Wproj, out);
}